// DistanceNetwork_37692632989824
// MI455X (gfx1250) — compile-verified
//
#include <hip/hip_runtime.h>
#include <hip/hip_bf16.h>

#define EPS 1e-10f
#define D_DIM 1024
#define BM 128
#define BN 128
#define KB 32
#define LDA 36   // padded LDS row stride in floats (bank-conflict-free, 16B-aligned rows)

typedef __attribute__((ext_vector_type(2))) float v2f;
typedef __attribute__((ext_vector_type(8))) float v8f;

__device__ __forceinline__ float wave_reduce_sum(float v) {
    #pragma unroll
    for (int off = 16; off > 0; off >>= 1)
        v += __shfl_xor(v, off, 32);
    return v;
}

// Async global -> LDS copy, 16B per lane, tracked by ASYNCcnt (CDNA5).
// lds_off: wave-relative LDS byte offset (low 32 bits of the generic pointer).
__device__ __forceinline__ void async_copy_b128(unsigned lds_off, const float* gptr) {
    asm volatile("global_load_async_to_lds_b128 %0, %1, off"
                 :: "v"(lds_off), "v"(gptr)
                 : "memory");
}

__device__ __forceinline__ void wait_async_all() {
    asm volatile("s_wait_asynccnt 0x0" ::: "memory");
}

// ---- pass 1: per-block partial sums of input_image^2 ----
__global__ void sumsq_partial_kernel(const float* __restrict__ in,
                                     float* __restrict__ partials, int n) {
    const float4* in4 = (const float4*)in;
    int n4 = n >> 2;
    float s = 0.f;
    for (int i = blockIdx.x * blockDim.x + threadIdx.x; i < n4;
         i += gridDim.x * blockDim.x) {
        float4 v = in4[i];
        s += v.x * v.x + v.y * v.y + v.z * v.z + v.w * v.w;
    }
    s = wave_reduce_sum(s);
    __shared__ float sh[8];
    int lane = threadIdx.x & 31, wv = threadIdx.x >> 5;
    if (lane == 0) sh[wv] = s;
    __syncthreads();
    if (wv == 0) {
        float t = (lane < (int)(blockDim.x >> 5)) ? sh[lane] : 0.f;
        t = wave_reduce_sum(t);
        if (lane == 0) partials[blockIdx.x] = t;
    }
}

// ---- pass 2: reduce partials to a single scalar (deterministic, no atomics) ----
__global__ void sumsq_final_kernel(const float* __restrict__ partials,
                                   float* __restrict__ out, int np) {
    float s = 0.f;
    for (int i = threadIdx.x; i < np; i += blockDim.x) s += partials[i];
    s = wave_reduce_sum(s);
    __shared__ float sh[8];
    int lane = threadIdx.x & 31, wv = threadIdx.x >> 5;
    if (lane == 0) sh[wv] = s;
    __syncthreads();
    if (threadIdx.x == 0) {
        float t = 0.f;
        for (int i = 0; i < (int)(blockDim.x >> 5); ++i) t += sh[i];
        out[0] = t;
    }
}

// ---- per-support-row rsqrt(max(sumsq, EPS)): one wave per row ----
__global__ void support_mag_kernel(const float* __restrict__ support,
                                   float* __restrict__ smag, int nrows) {
    int wv = threadIdx.x >> 5, lane = threadIdx.x & 31;
    int row = blockIdx.x * 8 + wv;
    if (row >= nrows) return;
    const float4* r = (const float4*)(support + (size_t)row * D_DIM);
    float s = 0.f;
    #pragma unroll
    for (int i = 0; i < D_DIM / 4 / 32; ++i) {   // 8 float4s per lane
        float4 v = r[lane + 32 * i];
        s += v.x * v.x + v.y * v.y + v.z * v.z + v.w * v.w;
    }
    s = wave_reduce_sum(s);
    if (lane == 0) smag[row] = rsqrtf(fmaxf(s, EPS));
}

// ---- main GEMM: C[n,b] = sum_d S[n,d]*I[b,d], scaled by smag[n]*imag ----
__global__ __launch_bounds__(256)
void cosine_gemm_kernel(const float* __restrict__ support,   // [N, D]
                        const float* __restrict__ input,     // [B, D]
                        const float* __restrict__ smag,      // [N]
                        const float* __restrict__ sumsq_in,  // [1]
                        float* __restrict__ out,             // [N, B]
                        int Bcols) {
    __shared__ float As[2][BM * LDA];
    __shared__ float Bs[2][BN * LDA];

    const int tid  = threadIdx.x;
    const int lane = tid & 31;
    const int wv   = tid >> 5;     // 0..7
    const int wm   = wv & 3;       // 32-row band within the 128-row block tile
    const int wn   = wv >> 2;      // 64-col band within the 128-col block tile

    const int m0 = blockIdx.y * BM;
    const int n0 = blockIdx.x * BN;

    // cooperative tile load mapping: each thread moves 4 float4s per matrix per slab
    const int lrow = tid >> 3;         // 0..31
    const int lcol = (tid & 7) * 4;    // 0,4,...,28

    const int halfk = lane >> 4;       // K pair / M+8 selector
    const int l16   = lane & 15;

    const int aBase = (wm * 32 + l16) * LDA + halfk * 2;
    const int bBase = (wn * 64 + l16) * LDA + halfk * 2;

    v8f acc[2][4] = {};

    // Stage one K-slab asynchronously into LDS buffer `buf`.
    auto stage = [&](int buf, int k0) {
        #pragma unroll
        for (int i = 0; i < 4; ++i) {
            int r = lrow + 32 * i;
            async_copy_b128((unsigned)(uintptr_t)&As[buf][r * LDA + lcol],
                            support + (size_t)(m0 + r) * D_DIM + k0 + lcol);
            async_copy_b128((unsigned)(uintptr_t)&Bs[buf][r * LDA + lcol],
                            input + (size_t)(n0 + r) * D_DIM + k0 + lcol);
        }
    };

    constexpr int NSLAB = D_DIM / KB;   // 32

    // prologue: fill buffer 0
    stage(0, 0);
    wait_async_all();
    __syncthreads();

    for (int kt = 0; kt < NSLAB; ++kt) {
        const int cur = kt & 1;
        // prefetch next slab into the other buffer while we compute
        if (kt + 1 < NSLAB) stage(cur ^ 1, (kt + 1) * KB);

        #pragma unroll
        for (int k4 = 0; k4 < KB; k4 += 4) {
            v2f a[2], b[4];
            #pragma unroll
            for (int tm = 0; tm < 2; ++tm)
                a[tm] = *(const v2f*)(&As[cur][aBase + tm * 16 * LDA + k4]);
            #pragma unroll
            for (int tn = 0; tn < 4; ++tn)
                b[tn] = *(const v2f*)(&Bs[cur][bBase + tn * 16 * LDA + k4]);
            #pragma unroll
            for (int tm = 0; tm < 2; ++tm)
                #pragma unroll
                for (int tn = 0; tn < 4; ++tn)
                    acc[tm][tn] = __builtin_amdgcn_wmma_f32_16x16x4_f32(
                        false, a[tm], false, b[tn],
                        (short)0, acc[tm][tn], false, false);
        }

        // next slab must have fully landed in LDS (all waves) before use
        wait_async_all();
        __syncthreads();
    }

    // epilogue: scale by per-row support magnitude and the global input magnitude
    const float imag = rsqrtf(fmaxf(sumsq_in[0], EPS));
    #pragma unroll
    for (int tm = 0; tm < 2; ++tm) {
        float sm[8];
        #pragma unroll
        for (int v = 0; v < 8; ++v)
            sm[v] = smag[m0 + wm * 32 + tm * 16 + v + 8 * halfk];
        #pragma unroll
        for (int tn = 0; tn < 4; ++tn) {
            int colg = n0 + wn * 64 + tn * 16 + l16;
            #pragma unroll
            for (int v = 0; v < 8; ++v) {
                int rowg = m0 + wm * 32 + tm * 16 + v + 8 * halfk;
                out[(size_t)rowg * Bcols + colg] = acc[tm][tn][v] * sm[v] * imag;
            }
        }
    }
}

extern "C" void kernel_launch(void* const* d_in, const int* in_sizes, int n_in,
                              void* d_out, int out_size, void* d_ws, size_t ws_size,
                              hipStream_t stream) {
    const float* support = (const float*)d_in[0];   // [8192, 1024] f32
    const float* input   = (const float*)d_in[1];   // [2048, 1024] f32
    float* out = (float*)d_out;                     // [8192, 2048] f32
    float* ws  = (float*)d_ws;

    const int D     = D_DIM;
    const int Nrows = in_sizes[0] / D;   // 8192
    const int Bcols = in_sizes[1] / D;   // 2048

    float* sumsq    = ws;          // [1]
    float* partials = ws + 64;     // [512]
    float* smag     = ws + 1024;   // [Nrows]

    sumsq_partial_kernel<<<512, 256, 0, stream>>>(input, partials, in_sizes[1]);
    sumsq_final_kernel<<<1, 256, 0, stream>>>(partials, sumsq, 512);
    support_mag_kernel<<<Nrows / 8, 256, 0, stream>>>(support, smag, Nrows);

    dim3 grid(Bcols / BN, Nrows / BM);
    cosine_gemm_kernel<<<grid, 256, 0, stream>>>(support, input, smag, sumsq, out, Bcols);
}